// GAT_36223754174747
// MI455X (gfx1250) — compile-verified
//
#include <hip/hip_runtime.h>
#include <hip/hip_bf16.h>

typedef float v2f __attribute__((ext_vector_type(2)));
typedef float v8f __attribute__((ext_vector_type(8)));

#define IN_F   128
#define OUT_F  64
#define NEG_SLOPE 0.2f

// ---------------------------------------------------------------------------
// Kernel 0: init out = bias (atomics accumulate on top), denom = 0, m = -inf
// ---------------------------------------------------------------------------
__global__ void gat_init_kernel(float* __restrict__ out,
                                const float* __restrict__ bias,
                                float* __restrict__ denom,
                                int* __restrict__ m_bits,
                                int NT) {
    int i = blockIdx.x * blockDim.x + threadIdx.x;
    int total = NT * OUT_F;
    if (i < total) out[i] = bias[i & (OUT_F - 1)];
    if (i < NT) {
        denom[i] = 0.0f;
        m_bits[i] = (int)0xFF800000u;   // -inf bit pattern
    }
}

// ---------------------------------------------------------------------------
// Kernel 1: x_proj = x @ W via V_WMMA_F32_16X16X4_F32, fused a_src/a_dst.
// 128 threads = 4 waves; each wave computes a 16x64 tile (4 accumulators).
//
// W is staged in LDS k-pair interleaved: Wp[kp*128 + col*2 + j] = W[2kp+j][col]
// so each B fragment {W[kk][col], W[kk+1][col]} is one contiguous ds_load_b64
// that lands directly in the even-aligned VGPR pair WMMA consumes.
// ---------------------------------------------------------------------------
__global__ __launch_bounds__(128) void gat_proj_kernel(
    const float* __restrict__ x, const float* __restrict__ W,
    const float* __restrict__ att_src, const float* __restrict__ att_dst,
    float* __restrict__ xproj, float* __restrict__ a_src,
    float* __restrict__ a_dst, int N, int NT) {

    __shared__ float Wp[(IN_F / 2) * OUT_F * 2];  // 32 KB, k-pair interleaved
    __shared__ float Alds[4][16 * IN_F];          // 32 KB: per-wave 16x128 A

    const int tid  = threadIdx.x;
    const int wave = tid >> 5;
    const int lane = tid & 31;
    const int ln   = lane & 15;   // col (B/D) / row (A) within 16
    const int hi   = lane >> 4;   // 0: K=k,k+1  1: K=k+2,k+3

    // Cooperative load + repack of W into k-pair interleaved LDS layout.
    for (int i = tid; i < IN_F * OUT_F; i += 128) {
        const int kk  = i >> 6;        // 0..127
        const int col = i & 63;        // 0..63
        Wp[(kk >> 1) * (OUT_F * 2) + col * 2 + (kk & 1)] = W[i];
    }

    const int rowBase = (blockIdx.x * 4 + wave) * 16;

    // Stage this wave's 16x128 A tile into LDS with coalesced float4 loads.
    float* aw = &Alds[wave][0];
    for (int r = 0; r < 16; ++r) {
        int row = rowBase + r;
        if (row >= N) row = N - 1;               // clamp; stores are guarded
        const float4 v = *(const float4*)(x + (size_t)row * IN_F + lane * 4);
        *(float4*)(aw + r * IN_F + lane * 4) = v;
    }
    __syncthreads();

    v8f acc0 = {}, acc1 = {}, acc2 = {}, acc3 = {};

    #pragma unroll
    for (int k = 0; k < IN_F; k += 4) {
        // A fragment (16x4 f32): lane holds row=ln, K = k+2*hi, k+2*hi+1
        v2f a = *(const v2f*)(aw + ln * IN_F + k + 2 * hi);
        // B fragments: contiguous b64 per tile; tiles 128 bytes apart
        const float* bb = Wp + (size_t)((k >> 1) + hi) * (OUT_F * 2) + ln * 2;
        v2f b0 = *(const v2f*)(bb);
        v2f b1 = *(const v2f*)(bb + 32);
        v2f b2 = *(const v2f*)(bb + 64);
        v2f b3 = *(const v2f*)(bb + 96);

        acc0 = __builtin_amdgcn_wmma_f32_16x16x4_f32(false, a, false, b0, (short)0, acc0, false, false);
        acc1 = __builtin_amdgcn_wmma_f32_16x16x4_f32(false, a, false, b1, (short)0, acc1, false, false);
        acc2 = __builtin_amdgcn_wmma_f32_16x16x4_f32(false, a, false, b2, (short)0, acc2, false, false);
        acc3 = __builtin_amdgcn_wmma_f32_16x16x4_f32(false, a, false, b3, (short)0, acc3, false, false);
    }

    // Attention vectors: lane ln covers columns ln, 16+ln, 32+ln, 48+ln.
    const float as0 = att_src[ln],      as1 = att_src[16 + ln];
    const float as2 = att_src[32 + ln], as3 = att_src[48 + ln];
    const float ad0 = att_dst[ln],      ad1 = att_dst[16 + ln];
    const float ad2 = att_dst[32 + ln], ad3 = att_dst[48 + ln];

    // C/D layout: VGPR i -> row (i + 8*hi), col ln (per 16-col tile).
    #pragma unroll
    for (int i = 0; i < 8; ++i) {
        float p_s = acc0[i] * as0 + acc1[i] * as1 + acc2[i] * as2 + acc3[i] * as3;
        float p_d = acc0[i] * ad0 + acc1[i] * ad1 + acc2[i] * ad2 + acc3[i] * ad3;
        // half-wave reduction: xor masks < 16 stay inside each 16-lane half
        #pragma unroll
        for (int off = 1; off < 16; off <<= 1) {
            p_s += __shfl_xor(p_s, off, 32);
            p_d += __shfl_xor(p_d, off, 32);
        }
        const int row = rowBase + i + 8 * hi;
        if (row < N) {
            float* xp = xproj + (size_t)row * OUT_F;
            xp[ln]      = acc0[i];
            xp[16 + ln] = acc1[i];
            xp[32 + ln] = acc2[i];
            xp[48 + ln] = acc3[i];
            if (ln == 0) {
                a_src[row] = p_s;
                if (row < NT) a_dst[row] = p_d;
            }
        }
    }
}

// ---------------------------------------------------------------------------
// Kernel 2: e = leaky_relu(a_src[src] + a_dst[dst]); segment-max via
// native int atomics (int-max for >=0, uint-min for <0).
// ---------------------------------------------------------------------------
__global__ void gat_edge_max_kernel(const int* __restrict__ es,
                                    const int* __restrict__ ed,
                                    const float* __restrict__ a_src,
                                    const float* __restrict__ a_dst,
                                    float* __restrict__ e_out,
                                    int* __restrict__ m_bits, int E) {
    int i = blockIdx.x * blockDim.x + threadIdx.x;
    if (i >= E) return;
    const int s = es[i], d = ed[i];
    float e = a_src[s] + a_dst[d];
    e = (e > 0.0f) ? e : NEG_SLOPE * e;
    e_out[i] = e;
    if (e >= 0.0f) atomicMax(&m_bits[d], __float_as_int(e));
    else           atomicMin((unsigned int*)&m_bits[d], __float_as_uint(e));
}

// ---------------------------------------------------------------------------
// Kernel 3: p = exp(e - m[dst]) in place; denom[dst] += p.
// ---------------------------------------------------------------------------
__global__ void gat_edge_exp_kernel(const int* __restrict__ ed,
                                    const int* __restrict__ m_bits,
                                    float* __restrict__ e_p,
                                    float* __restrict__ denom, int E) {
    int i = blockIdx.x * blockDim.x + threadIdx.x;
    if (i >= E) return;
    const int d = ed[i];
    const float m = __int_as_float(m_bits[d]);
    const float p = __expf(e_p[i] - m);
    e_p[i] = p;
    atomicAdd(&denom[d], p);
}

// ---------------------------------------------------------------------------
// Kernel 4: out[dst] += alpha * xproj[src].  One wave per edge; lane l
// handles columns 2l, 2l+1 (coalesced 256B float2 gather per edge row).
// ---------------------------------------------------------------------------
__global__ __launch_bounds__(256) void gat_edge_aggr_kernel(
    const int* __restrict__ es, const int* __restrict__ ed,
    const float* __restrict__ p, const float* __restrict__ denom,
    const float* __restrict__ xproj, float* __restrict__ out, int E) {
    const int gtid = blockIdx.x * blockDim.x + threadIdx.x;
    const int edge = gtid >> 5;
    const int lane = gtid & 31;
    if (edge >= E) return;
    const int s = es[edge], d = ed[edge];
    const float alpha = p[edge] / (denom[d] + 1e-16f);
    const float2 v = *(const float2*)(xproj + (size_t)s * OUT_F + lane * 2);
    float* o = out + (size_t)d * OUT_F + lane * 2;
    atomicAdd(o,     alpha * v.x);
    atomicAdd(o + 1, alpha * v.y);
}

// ---------------------------------------------------------------------------
// Launch: inputs = {x, edge_src, edge_dst, W, att_src, att_dst, bias, num_target}
// ---------------------------------------------------------------------------
extern "C" void kernel_launch(void* const* d_in, const int* in_sizes, int n_in,
                              void* d_out, int out_size, void* d_ws, size_t ws_size,
                              hipStream_t stream) {
    const float* x        = (const float*)d_in[0];
    const int*   edge_src = (const int*)d_in[1];
    const int*   edge_dst = (const int*)d_in[2];
    const float* W        = (const float*)d_in[3];
    const float* att_src  = (const float*)d_in[4];
    const float* att_dst  = (const float*)d_in[5];
    const float* bias     = (const float*)d_in[6];

    const int N  = in_sizes[0] / IN_F;     // 100000
    const int E  = in_sizes[1];            // 2000000
    const int NT = out_size / OUT_F;       // 20000 (num_target lives on device)

    float* out = (float*)d_out;

    // Workspace carve-up (~34.2 MB total)
    float* xproj  = (float*)d_ws;                 // N * 64
    float* a_src  = xproj + (size_t)N * OUT_F;    // N
    float* a_dst  = a_src + N;                    // NT
    float* e_p    = a_dst + NT;                   // E   (e, then p in place)
    float* denom  = e_p + E;                      // NT
    int*   m_bits = (int*)(denom + NT);           // NT

    // 0) init: out = bias, denom = 0, m = -inf
    {
        int total = NT * OUT_F;
        gat_init_kernel<<<(total + 255) / 256, 256, 0, stream>>>(
            out, bias, denom, m_bits, NT);
    }
    // 1) WMMA projection + fused attention dot products
    {
        int blocks = (N + 63) / 64;   // 64 rows per 4-wave block
        gat_proj_kernel<<<blocks, 128, 0, stream>>>(
            x, W, att_src, att_dst, xproj, a_src, a_dst, N, NT);
    }
    // 2) edge logits + segment max
    gat_edge_max_kernel<<<(E + 255) / 256, 256, 0, stream>>>(
        edge_src, edge_dst, a_src, a_dst, e_p, m_bits, E);
    // 3) exp + segment sum
    gat_edge_exp_kernel<<<(E + 255) / 256, 256, 0, stream>>>(
        edge_dst, m_bits, e_p, denom, E);
    // 4) weighted aggregation (wave per edge)
    {
        long long threads = (long long)E * 32;
        int blocks = (int)((threads + 255) / 256);
        gat_edge_aggr_kernel<<<blocks, 256, 0, stream>>>(
            edge_src, edge_dst, e_p, denom, xproj, out, E);
    }
}